// SoftMeanShift_953482740098
// MI455X (gfx1250) — compile-verified
//
#include <hip/hip_runtime.h>
#include <hip/hip_bf16.h>

// Soft mean-shift for MI455X (gfx1250), f32 WMMA 16x16x4 path.
// Shapes hardcoded from setup_inputs(): B=2, E=16, H=W=256 (N=65536), K=256.

#define E_DIM 16
#define K_MODES 256
#define NUM_B 2
#define NPIX 65536
#define CHUNK 512          // pixels per workgroup in the heavy kernels
#define CPAD (CHUNK + 1)   // LDS row pad for conflict-free column access
#define MPAD 17            // modes/w-tile row pad
#define NUM_ITERS 10
#define INV2BW2 2.0f       // 1/(2*0.5^2)
#define THR2 0.0625f       // (0.5*0.5)^2
#define MIN_CLUSTER 50

typedef __attribute__((ext_vector_type(2))) float v2f;
typedef __attribute__((ext_vector_type(8))) float v8f;

// ---- 16x16 dot tile: D[m][n] = sum_e modes[m][e]*emb[e][n], 4 chained
// V_WMMA_F32_16X16X4_F32. A 16x4 layout: V0 lanes0-15->K=0, lanes16-31->K=2;
// V1 -> K=1/K=3 (ISA 7.12.2). B 4x16 mirrors A; C/D: VGPR v <-> M=v(+8).
__device__ __forceinline__ v8f dot_tile(const float* mL, const float* eL,
                                        int m0, int nloc, int l16, int kA) {
  v8f c = {};
#pragma unroll
  for (int es = 0; es < 4; ++es) {
    v2f a, b;
    a.x = mL[(m0 + l16) * MPAD + es * 4 + kA];
    a.y = mL[(m0 + l16) * MPAD + es * 4 + kA + 1];
    b.x = eL[(es * 4 + kA) * CPAD + nloc + l16];
    b.y = eL[(es * 4 + kA + 1) * CPAD + nloc + l16];
    c = __builtin_amdgcn_wmma_f32_16x16x4_f32(false, a, false, b, (short)0, c,
                                              false, false);
  }
  return c;
}

__global__ void msInit(const float* __restrict__ emb,
                       const int* __restrict__ seeds,
                       float* __restrict__ modes, float* __restrict__ modeNorm,
                       int* __restrict__ keep, float* __restrict__ accW,
                       float* __restrict__ accWE, int* __restrict__ counts) {
  const int b = blockIdx.x, k = threadIdx.x;
  const int s = seeds[b * K_MODES + k];
  float nrm = 0.f;
#pragma unroll
  for (int e = 0; e < E_DIM; ++e) {
    float v = emb[((long)b * E_DIM + e) * NPIX + s];
    modes[((long)b * K_MODES + k) * E_DIM + e] = v;
    accWE[((long)b * K_MODES + k) * E_DIM + e] = 0.f;
    nrm += v * v;
  }
  modeNorm[b * K_MODES + k] = nrm;
  keep[b * K_MODES + k] = 1;
  accW[b * K_MODES + k] = 0.f;
  counts[b * (K_MODES + 1) + k] = 0;
  if (k == 0) counts[b * (K_MODES + 1) + K_MODES] = 0;
}

__global__ __launch_bounds__(256) void msAccumulate(
    const float* __restrict__ emb, const int* __restrict__ fgmask,
    const float* __restrict__ modes, const float* __restrict__ modeNorm,
    float* __restrict__ accW, float* __restrict__ accWE) {
  __shared__ float eL[E_DIM * CPAD];       // emb chunk [e][n], padded
  __shared__ float pnL[CHUNK];             // |e|^2, fg-folded (+1e5 => w=0)
  __shared__ float mL[K_MODES * MPAD];     // modes [k][e], padded
  __shared__ float mnL[K_MODES];           // |m|^2
  __shared__ float wT[8 * 16 * MPAD];      // per-wave w tile C->A relayout

  const int chunksPerB = NPIX / CHUNK;
  const int b = blockIdx.x / chunksPerB;
  const int nbase = (blockIdx.x % chunksPerB) * CHUNK;
  const int tid = threadIdx.x;

  for (int i = tid; i < E_DIM * CHUNK; i += 256) {
    int e = i / CHUNK, p = i % CHUNK;
    eL[e * CPAD + p] = emb[((long)b * E_DIM + e) * NPIX + nbase + p];
  }
  for (int i = tid; i < K_MODES * E_DIM; i += 256) {
    int k = i / E_DIM, e = i % E_DIM;
    mL[k * MPAD + e] = modes[((long)b * K_MODES + k) * E_DIM + e];
  }
  if (tid < K_MODES) mnL[tid] = modeNorm[b * K_MODES + tid];
  __syncthreads();
  for (int p = tid; p < CHUNK; p += 256) {
    float s = 0.f;
#pragma unroll
    for (int e = 0; e < E_DIM; ++e) { float v = eL[e * CPAD + p]; s += v * v; }
    pnL[p] = (fgmask[(long)b * NPIX + nbase + p] > 0) ? s : 1e5f;
  }
  __syncthreads();

  const int lane = tid & 31, wv = tid >> 5;
  const int half = lane >> 4, l16 = lane & 15, kA = half * 2;
  float* wTw = &wT[wv * 16 * MPAD];
  float* accWB = accW + b * K_MODES;
  float* accWEB = accWE + (long)b * K_MODES * E_DIM;
  const int waveBase = wv * (CHUNK / 8);  // 64 pixels per wave

  for (int mb = 0; mb < 16; ++mb) {
    const int m0 = mb * 16;
    float mn[8];
#pragma unroll
    for (int v = 0; v < 8; ++v) mn[v] = mnL[m0 + v + 8 * half];
    v8f cwe = {};
    float wsum[8] = {0.f, 0.f, 0.f, 0.f, 0.f, 0.f, 0.f, 0.f};
    for (int t = 0; t < (CHUNK / 8) / 16; ++t) {  // 4 pixel tiles
      const int nloc = waveBase + t * 16;
      v8f cd = dot_tile(mL, eL, m0, nloc, l16, kA);
      const float pn = pnL[nloc + l16];
#pragma unroll
      for (int v = 0; v < 8; ++v) {
        float d = fmaxf(mn[v] + pn - 2.f * cd[v], 0.f);
        float w = __expf(-INV2BW2 * d);
        wsum[v] += w;
        wTw[(v + 8 * half) * MPAD + l16] = w;   // C layout -> LDS
      }
      // second GEMM: D[m][e] += sum_n w[m][n]*emb[e][n]  (A=w, B=emb^T)
#pragma unroll
      for (int ks = 0; ks < 4; ++ks) {
        v2f aw, bw;
        aw.x = wTw[l16 * MPAD + ks * 4 + kA];
        aw.y = wTw[l16 * MPAD + ks * 4 + kA + 1];
        bw.x = eL[l16 * CPAD + nloc + ks * 4 + kA];
        bw.y = eL[l16 * CPAD + nloc + ks * 4 + kA + 1];
        cwe = __builtin_amdgcn_wmma_f32_16x16x4_f32(false, aw, false, bw,
                                                    (short)0, cwe, false, false);
      }
    }
#pragma unroll
    for (int v = 0; v < 8; ++v) {
      float s = wsum[v];
      s += __shfl_xor(s, 1); s += __shfl_xor(s, 2);
      s += __shfl_xor(s, 4); s += __shfl_xor(s, 8);
      if (l16 == 0) atomicAdd(&accWB[m0 + v + 8 * half], s);
    }
#pragma unroll
    for (int v = 0; v < 8; ++v)
      atomicAdd(&accWEB[(m0 + v + 8 * half) * E_DIM + l16], cwe[v]);
  }
}

__global__ void msUpdateMerge(float* __restrict__ modes,
                              float* __restrict__ modeNorm,
                              int* __restrict__ keep, float* __restrict__ accW,
                              float* __restrict__ accWE) {
  __shared__ float mLDS[K_MODES * MPAD];
  __shared__ int keepLDS[K_MODES];
  const int b = blockIdx.x, k = threadIdx.x;
  const float ws = fmaxf(accW[b * K_MODES + k], 1e-8f);
  float m[E_DIM];
  float nrm = 0.f;
#pragma unroll
  for (int e = 0; e < E_DIM; ++e) {
    long idx = ((long)b * K_MODES + k) * E_DIM + e;
    m[e] = accWE[idx] / ws;
    accWE[idx] = 0.f;                 // reset for next iteration
    modes[idx] = m[e];
    mLDS[k * MPAD + e] = m[e];
    nrm += m[e] * m[e];
  }
  modeNorm[b * K_MODES + k] = nrm;
  accW[b * K_MODES + k] = 0.f;
  keepLDS[k] = keep[b * K_MODES + k];
  __syncthreads();
  // greedy sequential merge scan (matches jax.lax.scan over rows)
  for (int i = 0; i < K_MODES - 1; ++i) {
    if (k > i && keepLDS[i] && keepLDS[k]) {
      float d = 0.f;
#pragma unroll
      for (int e = 0; e < E_DIM; ++e) {
        float t = mLDS[i * MPAD + e] - m[e];
        d += t * t;
      }
      if (d < THR2) keepLDS[k] = 0;
    }
    __syncthreads();
  }
  keep[b * K_MODES + k] = keepLDS[k];
}

__global__ __launch_bounds__(256) void msAssign(
    const float* __restrict__ emb, const int* __restrict__ fgmask,
    const float* __restrict__ modes, const float* __restrict__ modeNorm,
    const int* __restrict__ keep, float* __restrict__ labels,
    float* __restrict__ soft, int* __restrict__ counts) {
  __shared__ float eL[E_DIM * CPAD];
  __shared__ float pnL[CHUNK];
  __shared__ float mL[K_MODES * MPAD];
  __shared__ float mnL[K_MODES];  // keep-folded: dropped -> 1e31 (=> prob 0)

  const int chunksPerB = NPIX / CHUNK;
  const int b = blockIdx.x / chunksPerB;
  const int nbase = (blockIdx.x % chunksPerB) * CHUNK;
  const int tid = threadIdx.x;

  for (int i = tid; i < E_DIM * CHUNK; i += 256) {
    int e = i / CHUNK, p = i % CHUNK;
    eL[e * CPAD + p] = emb[((long)b * E_DIM + e) * NPIX + nbase + p];
  }
  for (int i = tid; i < K_MODES * E_DIM; i += 256) {
    int k = i / E_DIM, e = i % E_DIM;
    mL[k * MPAD + e] = modes[((long)b * K_MODES + k) * E_DIM + e];
  }
  if (tid < K_MODES)
    mnL[tid] = keep[b * K_MODES + tid] ? modeNorm[b * K_MODES + tid] : 1e31f;
  __syncthreads();
  for (int p = tid; p < CHUNK; p += 256) {
    float s = 0.f;
#pragma unroll
    for (int e = 0; e < E_DIM; ++e) { float v = eL[e * CPAD + p]; s += v * v; }
    pnL[p] = s;
  }
  __syncthreads();

  const int lane = tid & 31, wv = tid >> 5;
  const int half = lane >> 4, l16 = lane & 15, kA = half * 2;
  const int waveBase = wv * (CHUNK / 8);

  for (int t = 0; t < (CHUNK / 8) / 16; ++t) {
    const int nloc = waveBase + t * 16;
    const float pn = pnL[nloc + l16];
    const int n = nbase + nloc + l16;

    // sweep 1: min distance (== max logit) + argmin
    float best = 3.0e38f;
    int bestIdx = 0;
    for (int mb = 0; mb < 16; ++mb) {
      const int m0 = mb * 16;
      v8f cd = dot_tile(mL, eL, m0, nloc, l16, kA);
#pragma unroll
      for (int v = 0; v < 8; ++v) {
        const int m = m0 + v + 8 * half;
        float d = fmaxf(mnL[m] + pn - 2.f * cd[v], 0.f);
        if (d < best) { best = d; bestIdx = m; }  // strict < keeps first index
      }
    }
    {  // combine the two half-lanes covering the same pixel
      float ob = __shfl_xor(best, 16);
      int oi = __shfl_xor(bestIdx, 16);
      if (ob < best || (ob == best && oi < bestIdx)) { best = ob; bestIdx = oi; }
    }
    if (half == 0) {
      int hardL = (fgmask[(long)b * NPIX + n] > 0) ? (bestIdx + 1) : 0;
      labels[(long)b * NPIX + n] = (float)hardL;
      atomicAdd(&counts[b * (K_MODES + 1) + hardL], 1);
    }

    // sweep 2: exp(logit - maxlogit), partial sum, write unnormalized
    float S = 0.f;
    for (int mb = 0; mb < 16; ++mb) {
      const int m0 = mb * 16;
      v8f cd = dot_tile(mL, eL, m0, nloc, l16, kA);
#pragma unroll
      for (int v = 0; v < 8; ++v) {
        const int m = m0 + v + 8 * half;
        float d = fmaxf(mnL[m] + pn - 2.f * cd[v], 0.f);
        float p = __expf(-INV2BW2 * (d - best));
        S += p;
        soft[((long)b * K_MODES + m) * NPIX + n] = p;
      }
    }
    S += __shfl_xor(S, 16);
    const float invS = 1.f / S;  // S >= 1 (best mode contributes exp(0))

    // sweep 3: normalize in place (same wave re-reads what it just wrote)
    for (int mb = 0; mb < 16; ++mb) {
      const int m0 = mb * 16;
#pragma unroll
      for (int v = 0; v < 8; ++v) {
        const int m = m0 + v + 8 * half;
        soft[((long)b * K_MODES + m) * NPIX + n] *= invS;
      }
    }
  }
}

__global__ void msFinalize(float* __restrict__ labels,
                           const int* __restrict__ counts) {
  const long idx = (long)blockIdx.x * blockDim.x + threadIdx.x;
  if (idx >= (long)NUM_B * NPIX) return;
  const int b = (int)(idx / NPIX);
  const int h = (int)labels[idx];
  if (h > 0 && counts[b * (K_MODES + 1) + h] < MIN_CLUSTER) labels[idx] = 0.f;
}

__global__ void msCenters(const float* __restrict__ modes,
                          const int* __restrict__ keep,
                          float* __restrict__ centers) {
  const int b = blockIdx.x, k = threadIdx.x;
  const int kp = keep[b * K_MODES + k];
#pragma unroll
  for (int e = 0; e < E_DIM; ++e) {
    long idx = ((long)b * K_MODES + k) * E_DIM + e;
    centers[idx] = kp ? modes[idx] : 0.f;
  }
}

extern "C" void kernel_launch(void* const* d_in, const int* in_sizes, int n_in,
                              void* d_out, int out_size, void* d_ws,
                              size_t ws_size, hipStream_t stream) {
  const float* emb = (const float*)d_in[0];
  const int* fg = (const int*)d_in[1];
  const int* seeds = (const int*)d_in[2];

  // workspace layout (floats): ~74 KB total
  float* ws = (float*)d_ws;
  float* modes = ws;                              // B*K*E = 8192
  float* modeNorm = modes + NUM_B * K_MODES * E_DIM;  // 512
  int* keep = (int*)(modeNorm + NUM_B * K_MODES);     // 512
  float* accW = (float*)(keep + NUM_B * K_MODES);     // 512
  float* accWE = accW + NUM_B * K_MODES;              // 8192
  int* counts = (int*)(accWE + NUM_B * K_MODES * E_DIM);  // B*(K+1)

  // outputs: labels [B,N] | soft [B,K,N] | centers [B,K,E]
  float* labels = (float*)d_out;
  float* soft = labels + (long)NUM_B * NPIX;
  float* centers = soft + (long)NUM_B * K_MODES * NPIX;

  msInit<<<dim3(NUM_B), dim3(K_MODES), 0, stream>>>(
      emb, seeds, modes, modeNorm, keep, accW, accWE, counts);

  const int nWG = NUM_B * (NPIX / CHUNK);  // 256 workgroups
  for (int it = 0; it < NUM_ITERS; ++it) {
    msAccumulate<<<dim3(nWG), dim3(256), 0, stream>>>(emb, fg, modes, modeNorm,
                                                      accW, accWE);
    msUpdateMerge<<<dim3(NUM_B), dim3(K_MODES), 0, stream>>>(
        modes, modeNorm, keep, accW, accWE);
  }

  msAssign<<<dim3(nWG), dim3(256), 0, stream>>>(emb, fg, modes, modeNorm, keep,
                                                labels, soft, counts);
  msFinalize<<<dim3((NUM_B * NPIX) / 256), dim3(256), 0, stream>>>(labels,
                                                                   counts);
  msCenters<<<dim3(NUM_B), dim3(K_MODES), 0, stream>>>(modes, keep, centers);
}